// Model_48601849921966
// MI455X (gfx1250) — compile-verified
//
#include <hip/hip_runtime.h>
#include <math.h>

typedef __attribute__((ext_vector_type(2))) float v2f;
typedef __attribute__((ext_vector_type(8))) float v8f;
typedef __attribute__((ext_vector_type(4))) int   v4i;
typedef __attribute__((address_space(1))) v4i* gv4i_p;  // global
typedef __attribute__((address_space(3))) v4i* lv4i_p;  // LDS

#define TM 64
#define TN 64
#define TK 16
#define APAD 20   // A LDS row pitch (floats): 80B = 16B-aligned rows for b128 async
#define BPAD 18   // B^T LDS row pitch (floats): even -> K-pairs 8B aligned

#if defined(__has_builtin)
#  if __has_builtin(__builtin_amdgcn_global_load_async_to_lds_b128)
#    define USE_ASYNC_LDS 1
#  endif
#  if __has_builtin(__builtin_amdgcn_s_wait_asynccnt)
#    define HAVE_WAIT_ASYNC 1
#  endif
#endif

__device__ __forceinline__ void wait_async_zero() {
#if defined(USE_ASYNC_LDS)
#  if defined(HAVE_WAIT_ASYNC)
  __builtin_amdgcn_s_wait_asynccnt(0);
#  else
  asm volatile("s_wait_asynccnt 0" ::: "memory");
#  endif
#endif
}

// ---------------------------------------------------------------------------
// Fused WMMA fp32 GEMM:  C = epilogue(A[M,K] @ B[K,N])
//   EPI==0 : C = relu(v - bias[n])       (GEMM1: pre-activation + shift)
//   EPI==1 : C = sigmoid(v + bias[0])    (GEMM2: output head)
// 128 threads = 4 wave32; block tile 64x64; wave tile 16x64 (4x v_wmma).
// A staged via GLOBAL_LOAD_ASYNC_TO_LDS_B128 (ASYNCcnt) when available;
// B staged transposed so each 4x16 B-fragment is a contiguous, 8B-aligned
// DWORD pair in LDS (single DS load, no repack movs).
// All dims are exact tile multiples => EXEC all-1s around WMMA.
// ---------------------------------------------------------------------------
template <int EPI>
__global__ __launch_bounds__(128) void gemm_wmma_f32(
    const float* __restrict__ A, const float* __restrict__ Bm,
    const float* __restrict__ bias, float* __restrict__ C,
    int M, int N, int K) {
  __shared__ float As[TM][APAD];    // A tile, row-major  [64 x 16(+4)]
  __shared__ float Bst[TN][BPAD];   // B tile, TRANSPOSED [64(N) x 16(+2)(K)]

  const int tid  = threadIdx.x;
  const int wave = tid >> 5;   // wave32: 0..3
  const int lane = tid & 31;
  const int m0 = blockIdx.y * TM;
  const int n0 = blockIdx.x * TN;

  const int row16 = lane & 15;          // M (A) or N (B/C) index within 16
  const int khalf = (lane >> 4) << 1;   // lanes 16..31 hold the K+2/K+3 half

  v8f acc[4] = {};   // 4 N-subtiles of 16x16 f32 accumulators (8 VGPRs each)

  for (int k0 = 0; k0 < K; k0 += TK) {
    // ---- stage A (64x16): async DMA to LDS if available -----------------
#if defined(USE_ASYNC_LDS)
#pragma unroll
    for (int it = 0; it < 2; ++it) {
      const int p  = it * 512 + tid * 4;     // 128 lanes x 16B = 512 floats
      const int ar = p >> 4, ac = p & 15;    // ac in {0,4,8,12} -> 16B aligned
      __builtin_amdgcn_global_load_async_to_lds_b128(
          (gv4i_p)(A + (size_t)(m0 + ar) * K + k0 + ac),
          (lv4i_p)&As[ar][ac],
          /*offset=*/0, /*cpol=*/0);
    }
#else
#pragma unroll
    for (int it = 0; it < 2; ++it) {
      const int p  = it * 512 + tid * 4;
      const int ar = p >> 4, ac = p & 15;
      const float4 av = *(const float4*)(A + (size_t)(m0 + ar) * K + k0 + ac);
      As[ar][ac + 0] = av.x; As[ar][ac + 1] = av.y;
      As[ar][ac + 2] = av.z; As[ar][ac + 3] = av.w;
    }
#endif
    // ---- stage B (16x64) transposed into Bst[n][k] ----------------------
#pragma unroll
    for (int it = 0; it < 2; ++it) {
      const int p  = it * 512 + tid * 4;
      const int br = p >> 6, bc = p & 63;
      const float4 bv = *(const float4*)(Bm + (size_t)(k0 + br) * N + n0 + bc);
      Bst[bc + 0][br] = bv.x; Bst[bc + 1][br] = bv.y;
      Bst[bc + 2][br] = bv.z; Bst[bc + 3][br] = bv.w;
    }
    // hint the next K-slab into cache (global_prefetch_b8)
    if (k0 + TK < K) {
      __builtin_prefetch(A + (size_t)(m0 + (tid >> 1)) * K + k0 + TK, 0, 1);
      __builtin_prefetch(Bm + (size_t)(k0 + TK + (tid & 15)) * N + n0, 0, 1);
    }
    wait_async_zero();
    __syncthreads();

    // ---- 4 WMMA K-steps over the 16-wide slab ---------------------------
#pragma unroll
    for (int kk = 0; kk < TK; kk += 4) {
      v2f a;  // 16x4 A-frag: lanes 0-15 {K,K+1}, lanes 16-31 {K+2,K+3}
      a.x = As[wave * 16 + row16][kk + khalf + 0];
      a.y = As[wave * 16 + row16][kk + khalf + 1];
#pragma unroll
      for (int t = 0; t < 4; ++t) {
        v2f b;  // 4x16 B-frag; contiguous K-pair in transposed LDS tile
        b.x = Bst[t * 16 + row16][kk + khalf + 0];
        b.y = Bst[t * 16 + row16][kk + khalf + 1];
        acc[t] = __builtin_amdgcn_wmma_f32_16x16x4_f32(
            /*neg_a=*/false, a, /*neg_b=*/false, b,
            /*c_mod=*/(short)0, acc[t], /*reuse_a=*/false, /*reuse_b=*/false);
      }
    }
    __syncthreads();
  }

  // ---- epilogue + store (C layout: VGPR v -> M=v / M=v+8 halves) --------
  const int mbase = m0 + wave * 16 + ((lane >> 4) << 3);
#pragma unroll
  for (int t = 0; t < 4; ++t) {
    const int col = n0 + t * 16 + row16;
#pragma unroll
    for (int v = 0; v < 8; ++v) {
      float val = acc[t][v];
      if (EPI == 0) {
        val -= bias[col];
        val = val > 0.0f ? val : 0.0f;              // relu(xW1 - b1)
      } else {
        val += bias[0];
        val = 1.0f / (1.0f + __expf(-val));         // sigmoid(. + b2)
      }
      C[(size_t)(mbase + v) * N + col] = val;
    }
  }
}

// ---------------------------------------------------------------------------
// Gamma(phi,1) sampling via Marsaglia-Tsang (+ boost for alpha<1), with a
// stateless counter-hash RNG; also emits phi = softplus(param) to d_out tail.
// ---------------------------------------------------------------------------
__device__ __forceinline__ unsigned pcg_hash(unsigned x) {
  x = x * 747796405u + 2891336453u;
  unsigned w = ((x >> ((x >> 28u) + 4u)) ^ x) * 277803737u;
  return (w >> 22u) ^ w;
}
__device__ __forceinline__ float u01(unsigned r) {
  return (float)(r >> 8) * (1.0f / 16777216.0f) + (0.5f / 16777216.0f);
}

__global__ __launch_bounds__(256) void gamma_kernel(
    const float* __restrict__ param, float* __restrict__ phi_out,
    float* __restrict__ gam, int H, int K) {
  const int idx = blockIdx.x * blockDim.x + threadIdx.x;  // (h,k) flattened
  const int h = idx / K;
  const int k = idx - h * K;
  const float p = param[h];
  const float phi = (p > 20.0f) ? p : log1pf(__expf(p));  // softplus
  if (k == 0) phi_out[h] = phi;

  const bool boost = phi < 1.0f;
  const float ap = boost ? phi + 1.0f : phi;
  const float d = ap - (1.0f / 3.0f);
  const float c = rsqrtf(9.0f * d);
  const unsigned seed = 0x9E3779B9u ^ (unsigned)idx * 2654435761u;

  float sample = d;  // fallback (mode-ish) if rejection loop exhausts
#pragma unroll 1
  for (int it = 0; it < 32; ++it) {
    const float u1 = u01(pcg_hash(seed + 3u * it + 1u));
    const float u2 = u01(pcg_hash(seed + 3u * it + 2u));
    const float u3 = u01(pcg_hash(seed + 3u * it + 3u));
    const float rad = sqrtf(-2.0f * __logf(u1));
    const float x = rad * __cosf(6.2831853f * u2);   // Box-Muller normal
    const float ocx = 1.0f + c * x;
    if (ocx <= 0.0f) continue;
    const float v = ocx * ocx * ocx;
    if (__logf(u3) < 0.5f * x * x + d - d * v + d * __logf(v)) {
      sample = d * v;
      break;
    }
  }
  if (boost) {
    const float u = u01(pcg_hash(seed ^ 0xABCD1234u));
    sample *= __powf(u, 1.0f / phi);   // gamma(a) = gamma(a+1) * U^(1/a)
  }
  gam[idx] = sample;
}

// per-column (k) reciprocal sums over h (S normalization)
__global__ __launch_bounds__(256) void colsum_kernel(
    const float* __restrict__ gam, float* __restrict__ colrecip, int H, int K) {
  __shared__ float red[256];
  const int k = blockIdx.x;
  float s = 0.0f;
  for (int h = threadIdx.x; h < H; h += 256) s += gam[(size_t)h * K + k];
  red[threadIdx.x] = s;
  __syncthreads();
  for (int off = 128; off > 0; off >>= 1) {
    if (threadIdx.x < off) red[threadIdx.x] += red[threadIdx.x + off];
    __syncthreads();
  }
  if (threadIdx.x == 0) colrecip[k] = 1.0f / red[0];
}

// Bmat[h,k] = gamma[h,k] / colsum[k] * W2[h]   (S folded with W2)
__global__ __launch_bounds__(256) void build_b_kernel(
    const float* __restrict__ gam, const float* __restrict__ colrecip,
    const float* __restrict__ W2, float* __restrict__ Bmat, int H, int K) {
  const int idx = blockIdx.x * blockDim.x + threadIdx.x;
  const int h = idx / K;
  const int k = idx - h * K;
  Bmat[idx] = gam[idx] * colrecip[k] * W2[h];
}

// ---------------------------------------------------------------------------
extern "C" void kernel_launch(void* const* d_in, const int* in_sizes, int n_in,
                              void* d_out, int out_size, void* d_ws, size_t ws_size,
                              hipStream_t stream) {
  constexpr int B = 512, D_IN = 1024, H = 1024, K = 128;

  const float* x     = (const float*)d_in[0];
  const float* W1    = (const float*)d_in[1];
  const float* b1    = (const float*)d_in[2];
  const float* W2    = (const float*)d_in[3];
  const float* b2    = (const float*)d_in[4];
  const float* param = (const float*)d_in[5];

  float* out     = (float*)d_out;            // [B,K] sigmoid output
  float* phi_out = out + (size_t)B * K;      // [H] phi (tuple element 2)

  float* ws       = (float*)d_ws;
  float* Aact     = ws;                      // [B,H]  relu(xW1 - b1)
  float* gam      = Aact + (size_t)B * H;    // [H,K]  gamma samples
  float* Bmat     = gam + (size_t)H * K;     // [H,K]  S * W2
  float* colrecip = Bmat + (size_t)H * K;    // [K]

  // 1) gamma samples + phi
  gamma_kernel<<<(H * K) / 256, 256, 0, stream>>>(param, phi_out, gam, H, K);
  // 2) column normalizers
  colsum_kernel<<<K, 256, 0, stream>>>(gam, colrecip, H, K);
  // 3) fold S with W2
  build_b_kernel<<<(H * K) / 256, 256, 0, stream>>>(gam, colrecip, W2, Bmat, H, K);
  // 4) GEMM1: Aact = relu(x @ W1 - b1)     [512x1024]x[1024x1024]
  gemm_wmma_f32<0><<<dim3(H / TN, B / TM), 128, 0, stream>>>(
      x, W1, b1, Aact, B, H, D_IN);
  // 5) GEMM2: out = sigmoid(Aact @ Bmat + b2)   [512x1024]x[1024x128]
  gemm_wmma_f32<1><<<dim3(K / TN, B / TM), 128, 0, stream>>>(
      Aact, Bmat, b2, out, B, K, H);
}